// GNNNet_82197084111332
// MI455X (gfx1250) — compile-verified
//
#include <hip/hip_runtime.h>
#include <math.h>

typedef __attribute__((ext_vector_type(2))) float v2f;
typedef __attribute__((ext_vector_type(8))) float v8f;

#define NFEAT   128
#define NGRAPH  200

// ---------------------------------------------------------------- utilities

__global__ void k_fill(float* __restrict__ p, float v, long n) {
  long i = (long)blockIdx.x * blockDim.x + threadIdx.x;
  if (i < n) p[i] = v;
}

__device__ __forceinline__ void edge_sd(const int* __restrict__ ei, int E, int e,
                                        int& s, int& d) {
  if (e < E) { s = ei[e]; d = ei[E + e]; }
  else       { s = e - E; d = e - E; }          // self loop
}

__device__ __forceinline__ void atomicMaxF(float* addr, float val) {
  int* ia = (int*)addr;
  int old = __float_as_int(*addr);
  while (__int_as_float(old) < val) {
    int assumed = old;
    old = atomicCAS(ia, assumed, __float_as_int(val));
    if (old == assumed) break;
  }
}

// ------------------------------------------------------- fp32 WMMA GEMM
// C[M,N] = A[M,K] @ B[K,N] (+ bias[N] if non-null).
// REQUIRES: M % 16 == 0, N % 16 == 0 (true for all call sites here).
// One wave32 computes one 16x16 output tile with V_WMMA_F32_16X16X4_F32.
// A-frag (ISA 7.12.2, 32-bit A 16x4): lanes 0-15 hold K={0,1}, lanes 16-31
// hold K={2,3}; B/C/D: row striped across lanes within a VGPR.
// Main K loop is guard-free (K-tail peeled) so loads stay unpredicated.
__global__ void k_gemm_wmma(const float* __restrict__ A,
                            const float* __restrict__ B,
                            const float* __restrict__ bias,
                            float* __restrict__ C,
                            int M, int K, int N) {
  int wave = threadIdx.x >> 5;
  int lane = threadIdx.x & 31;
  int half = lane >> 4;
  int l15  = lane & 15;
  int nt   = N >> 4;
  int tile = blockIdx.x * (blockDim.x >> 5) + wave;
  int mt   = tile / nt;
  int ntl  = tile % nt;
  int m0 = mt * 16;
  int n0 = ntl * 16;
  if (m0 >= M) return;                          // wave-uniform exit

  int row = m0 + l15;
  int col = n0 + l15;
  const float* Arow = A + row * K + 2 * half;   // lane's K-pair base
  const float* Bcol = B + col;

  v8f acc = {};
  const int Kmain = K & ~3;
#pragma unroll 4
  for (int k = 0; k < Kmain; k += 4) {
    v2f a = *(const v2f*)(Arow + k);            // 8B-aligned: K,k,2*half all even
    int kb = k + 2 * half;
    v2f b;
    b.x = Bcol[kb * N];
    b.y = Bcol[(kb + 1) * N];
    acc = __builtin_amdgcn_wmma_f32_16x16x4_f32(
        false, a, false, b, (short)0, acc, false, false);
  }
  if (Kmain < K) {                              // peeled tail (K = 70 / 78)
    int ka = Kmain + 2 * half;
    v2f a, b;
    a.x = (ka     < K) ? Arow[Kmain    ] : 0.f;
    a.y = (ka + 1 < K) ? Arow[Kmain + 1] : 0.f;
    b.x = (ka     < K) ? Bcol[ka * N]       : 0.f;
    b.y = (ka + 1 < K) ? Bcol[(ka + 1) * N] : 0.f;
    acc = __builtin_amdgcn_wmma_f32_16x16x4_f32(
        false, a, false, b, (short)0, acc, false, false);
  }

  float bb = bias ? bias[col] : 0.f;
#pragma unroll
  for (int r = 0; r < 8; ++r) {
    int rm = m0 + r + 8 * half;
    C[rm * N + col] = acc[r] + bb;
  }
}

// -------------------------------------------------------- GAT attention

// per-(node,head) attention logits: a_src/a_dst = <h[n,h,:], att[h,:]>
__global__ void k_att(const float* __restrict__ Hb,
                      const float* __restrict__ att_src,
                      const float* __restrict__ att_dst,
                      float* __restrict__ asrc, float* __restrict__ adst,
                      int N, int H, int C) {
  long i = (long)blockIdx.x * blockDim.x + threadIdx.x;
  if (i >= (long)N * H) return;
  int n = (int)(i / H), h = (int)(i % H);
  const float* hv = Hb + (long)n * NFEAT + h * C;
  float s = 0.f, d = 0.f;
  for (int c = 0; c < C; ++c) {
    float v = hv[c];
    s += v * att_src[h * C + c];
    d += v * att_dst[h * C + c];
  }
  asrc[i] = s;
  adst[i] = d;
}

__global__ void k_edge_max(const float* __restrict__ asrc,
                           const float* __restrict__ adst,
                           const int* __restrict__ ei, int E, long EtotH, int H,
                           float* __restrict__ m) {
  long i = (long)blockIdx.x * blockDim.x + threadIdx.x;
  if (i >= EtotH) return;
  int e = (int)(i / H), h = (int)(i % H);
  int s, d; edge_sd(ei, E, e, s, d);
  float v = asrc[(long)s * H + h] + adst[(long)d * H + h];
  v = v > 0.f ? v : 0.2f * v;                  // leaky_relu(0.2)
  atomicMaxF(&m[(long)d * H + h], v);
}

__global__ void k_edge_exp(const float* __restrict__ asrc,
                           const float* __restrict__ adst,
                           const int* __restrict__ ei, int E, long EtotH, int H,
                           const float* __restrict__ m,
                           float* __restrict__ den, float* __restrict__ exb) {
  long i = (long)blockIdx.x * blockDim.x + threadIdx.x;
  if (i >= EtotH) return;
  int e = (int)(i / H), h = (int)(i % H);
  int s, d; edge_sd(ei, E, e, s, d);
  float v = asrc[(long)s * H + h] + adst[(long)d * H + h];
  v = v > 0.f ? v : 0.2f * v;
  float ex = expf(v - m[(long)d * H + h]);
  exb[i] = ex;
  atomicAdd(&den[(long)d * H + h], ex);
}

// weighted scatter: gat[dst, j] += h[src, j] * alpha(e, head(j))
__global__ void k_edge_agg(const float* __restrict__ Hb,
                           const float* __restrict__ exb,
                           const float* __restrict__ den,
                           const int* __restrict__ ei, int E, long EtotF,
                           int H, int cshift, float* __restrict__ gat) {
  long i = (long)blockIdx.x * blockDim.x + threadIdx.x;
  if (i >= EtotF) return;
  int e = (int)(i >> 7);
  int j = (int)(i & (NFEAT - 1));
  int h = j >> cshift;
  int s, d; edge_sd(ei, E, e, s, d);
  float alpha = exb[(long)e * H + h] / (den[(long)d * H + h] + 1e-16f);
  atomicAdd(&gat[(long)d * NFEAT + j], Hb[(long)s * NFEAT + j] * alpha);
}

// X = elu(gat + gat_bias)   (gat already holds lin(x)+lin_bias + messages)
__global__ void k_combine_elu(const float* __restrict__ gat,
                              const float* __restrict__ gbias,
                              float* __restrict__ X, long NF) {
  long i = (long)blockIdx.x * blockDim.x + threadIdx.x;
  if (i >= NF) return;
  int j = (int)(i & (NFEAT - 1));
  float v = gat[i] + gbias[j];
  X[i] = v > 0.f ? v : expm1f(v);
}

// ------------------------------------------------------------ batch norm

__global__ void k_bn_stats(const float* __restrict__ X,
                           float* __restrict__ stats, int N) {
  int j = threadIdx.x;                         // 128 features
  long r0 = (long)blockIdx.x * 256;
  long r1 = r0 + 256; if (r1 > N) r1 = N;
  float s = 0.f, q = 0.f;
  for (long r = r0; r < r1; ++r) {
    float v = X[r * NFEAT + j];
    s += v; q += v * v;
  }
  atomicAdd(&stats[j], s);
  atomicAdd(&stats[NFEAT + j], q);
}

__global__ void k_bn_apply(float* __restrict__ X,
                           const float* __restrict__ stats,
                           const float* __restrict__ gamma,
                           const float* __restrict__ beta, int N) {
  long i = (long)blockIdx.x * blockDim.x + threadIdx.x;
  if (i >= (long)N * NFEAT) return;
  int j = (int)(i & (NFEAT - 1));
  float inv = 1.f / (float)N;
  float mu  = stats[j] * inv;
  float var = stats[NFEAT + j] * inv - mu * mu;
  X[i] = gamma[j] * (X[i] - mu) * rsqrtf(var + 1e-5f) + beta[j];
}

// ------------------------------------------------------------ mean pool

__global__ void k_pool_accum(const float* __restrict__ X,
                             const int* __restrict__ batch,
                             float* __restrict__ out, float* __restrict__ cnt,
                             int N) {
  long i = (long)blockIdx.x * blockDim.x + threadIdx.x;
  if (i >= (long)N * NFEAT) return;
  int n = (int)(i >> 7);
  int j = (int)(i & (NFEAT - 1));
  int g = batch[n];
  atomicAdd(&out[(long)g * NFEAT + j], X[i]);
  if (j == 0) atomicAdd(&cnt[g], 1.f);
}

__global__ void k_pool_div(float* __restrict__ out, const float* __restrict__ cnt) {
  long i = (long)blockIdx.x * blockDim.x + threadIdx.x;
  if (i >= (long)NGRAPH * NFEAT) return;
  float c = cnt[i >> 7];
  out[i] /= (c > 1.f ? c : 1.f);
}

// ---------------------------------------------------------------- driver

struct GatP { const float *W, *att_src, *att_dst, *bias; };
struct LinP { const float *W, *b; };
struct BnP  { const float *gamma, *beta; };

struct Arena {
  float *XA, *Hb, *GAT, *asrc, *adst, *m, *den, *exb, *stats, *cnt;
};

static inline unsigned gb(long n, int b) { return (unsigned)((n + b - 1) / b); }

static void run_branch(hipStream_t st, const float* x_raw, int Kin, int N,
                       const int* ei, int E, const int* batch,
                       const GatP cv[3], const LinP ln[3], const BnP bn[3],
                       float* out_seg, const Arena& a) {
  const int Etot = E + N;
  for (int l = 0; l < 3; ++l) {
    const int H = (l < 2) ? 4 : 1;
    const int C = (l < 2) ? 32 : 128;
    const int cshift = (l < 2) ? 5 : 7;
    const float* Xin = (l == 0) ? x_raw : a.XA;
    const int K = (l == 0) ? Kin : NFEAT;

    // GEMMs: Hb = Xin @ Wc ; GAT = Xin @ Wl + bl  (lin fused into accumulator)
    long tiles = (long)(N / 16) * (NFEAT / 16);
    k_gemm_wmma<<<gb(tiles, 8), 256, 0, st>>>(Xin, cv[l].W, nullptr, a.Hb, N, K, NFEAT);
    k_gemm_wmma<<<gb(tiles, 8), 256, 0, st>>>(Xin, ln[l].W, ln[l].b, a.GAT, N, K, NFEAT);

    // attention softmax over incoming edges (+ self loops)
    long NH = (long)N * H, EH = (long)Etot * H, EF = (long)Etot * NFEAT;
    k_fill<<<gb(NH, 256), 256, 0, st>>>(a.m, -1e30f, NH);
    k_fill<<<gb(NH, 256), 256, 0, st>>>(a.den, 0.f, NH);
    k_att<<<gb(NH, 256), 256, 0, st>>>(a.Hb, cv[l].att_src, cv[l].att_dst,
                                       a.asrc, a.adst, N, H, C);
    k_edge_max<<<gb(EH, 256), 256, 0, st>>>(a.asrc, a.adst, ei, E, EH, H, a.m);
    k_edge_exp<<<gb(EH, 256), 256, 0, st>>>(a.asrc, a.adst, ei, E, EH, H,
                                            a.m, a.den, a.exb);
    k_edge_agg<<<gb(EF, 256), 256, 0, st>>>(a.Hb, a.exb, a.den, ei, E, EF,
                                            H, cshift, a.GAT);

    // elu(gat + lin) then batch norm (in place on XA)
    long NF = (long)N * NFEAT;
    k_combine_elu<<<gb(NF, 256), 256, 0, st>>>(a.GAT, cv[l].bias, a.XA, NF);
    k_fill<<<1, 256, 0, st>>>(a.stats, 0.f, 2 * NFEAT);
    k_bn_stats<<<gb(N, 256), NFEAT, 0, st>>>(a.XA, a.stats, N);
    k_bn_apply<<<gb(NF, 256), 256, 0, st>>>(a.XA, a.stats, bn[l].gamma,
                                            bn[l].beta, N);
  }
  // global mean pool
  k_fill<<<1, 256, 0, st>>>(a.cnt, 0.f, NGRAPH);
  k_pool_accum<<<gb((long)N * NFEAT, 256), 256, 0, st>>>(a.XA, batch, out_seg,
                                                         a.cnt, N);
  k_pool_div<<<gb((long)NGRAPH * NFEAT, 256), 256, 0, st>>>(out_seg, a.cnt);
}

extern "C" void kernel_launch(void* const* d_in, const int* in_sizes, int n_in,
                              void* d_out, int out_size, void* d_ws, size_t ws_size,
                              hipStream_t stream) {
  (void)in_sizes; (void)n_in; (void)ws_size;
  auto f  = [&](int i) { return (const float*)d_in[i]; };
  auto ii = [&](int i) { return (const int*)d_in[i]; };

  // d_in layout: top-level dict insertion order, params pytree-flattened
  // (sorted keys: W < att_dst < att_src < bias ; beta < gamma).
  const float* mol_x = f(0);
  const float* pro_x = f(1);
  BnP bn_mol[3], bn_pro[3];
  for (int l = 0; l < 3; ++l) {
    bn_mol[l].beta = f(2 + 2 * l);  bn_mol[l].gamma = f(3 + 2 * l);
    bn_pro[l].beta = f(8 + 2 * l);  bn_pro[l].gamma = f(9 + 2 * l);
  }
  GatP cv_mol[3], cv_pro[3];
  LinP ln_mol[3], ln_pro[3];
  for (int l = 0; l < 3; ++l) {
    int b = 14 + 4 * l;                 // mol_conv{1,2,3}
    cv_mol[l] = {f(b), f(b + 2), f(b + 1), f(b + 3)};   // W, att_src, att_dst, bias
    int c = 26 + 2 * l;                 // mol_lin{1,2,3}
    ln_mol[l] = {f(c), f(c + 1)};
    int d = 32 + 4 * l;                 // pro_conv{1,2,3}
    cv_pro[l] = {f(d), f(d + 2), f(d + 1), f(d + 3)};
    int e = 44 + 2 * l;                 // pro_lin{1,2,3}
    ln_pro[l] = {f(e), f(e + 1)};
  }
  const int* mol_ei    = ii(50);
  const int* mol_batch = ii(51);
  const int* pro_ei    = ii(52);
  const int* pro_batch = ii(53);

  // scratch arena (sized for the protein branch, reused by both)
  const long NMAX = 100000, ETOT = 1100000, HM = 4;
  float* w = (float*)d_ws;
  size_t o = 0;
  Arena a;
  a.XA    = w + o; o += (size_t)NMAX * NFEAT;
  a.Hb    = w + o; o += (size_t)NMAX * NFEAT;
  a.GAT   = w + o; o += (size_t)NMAX * NFEAT;
  a.asrc  = w + o; o += (size_t)NMAX * HM;
  a.adst  = w + o; o += (size_t)NMAX * HM;
  a.m     = w + o; o += (size_t)NMAX * HM;
  a.den   = w + o; o += (size_t)NMAX * HM;
  a.exb   = w + o; o += (size_t)ETOT * HM;
  a.stats = w + o; o += 2 * NFEAT;
  a.cnt   = w + o; o += NGRAPH;

  float* out = (float*)d_out;
  k_fill<<<gb(out_size, 256), 256, 0, stream>>>(out, 0.f, out_size);

  // reference returns (xt, x): protein pooled first, then molecule pooled
  run_branch(stream, pro_x, 70, 100000, pro_ei, 1000000, pro_batch,
             cv_pro, ln_pro, bn_pro, out, a);
  run_branch(stream, mol_x, 78, 8000, mol_ei, 32000, mol_batch,
             cv_mol, ln_mol, bn_mol, out + NGRAPH * NFEAT, a);
}